// StructuredVariational_86981677678682
// MI455X (gfx1250) — compile-verified
//
#include <hip/hip_runtime.h>
#include <math.h>
#include <stdint.h>

// Problem constants (match reference)
#define DZ      64
#define DY      16
#define NBLK    512
#define NSAMPLE 128
#define NY      8192          // NBLK*DY
#define DTOT    8256          // DZ + NY
#define JITTERF 1e-4f
#define NB      128           // Cholesky panel width
#define LP      132           // LDS row stride (floats): 16B-aligned, conflict-free

typedef __attribute__((ext_vector_type(2))) float v2f;
typedef __attribute__((ext_vector_type(8))) float v8f;

// ---------------------------------------------------------------------------
// CDNA5 WMMA f32 16x16x4.
// A: 16x4 (2 VGPRs): (m,k) -> lane = m + 16*(k>=2), vgpr = k&1
// B: 4x16 (2 VGPRs): (k,n) -> lane = n + 16*(k>=2), vgpr = k&1
// C/D: 16x16 (8 VGPRs): (m,n) -> lane = n + 16*(m>=8), vgpr = m&7
// ---------------------------------------------------------------------------
__device__ __forceinline__ v8f wmma4(v2f a, v2f b, v8f c) {
  return __builtin_amdgcn_wmma_f32_16x16x4_f32(
      false, a, false, b, (short)0, c, false, false);
}

// ---------------------------------------------------------------------------
// Generic NT-column-tile NT GEMM: acc = A(MxK row-major) * Bt(NxK row-major)^T
// mode 0: C = acc ; 1: C -= acc ; 2: C += acc ; 3: C = acc + bias[col]
// One wave computes a 16 x (NT*16) strip; register double-buffered k-loop.
// M%16==0, N%(NT*16)==0, K%4==0, K>=8.
// NOTE: no __restrict__ — TRSM runs with A == C in place (wave owns the full
// N-wide strip of its rows, reads complete before the epilogue writes).
// ---------------------------------------------------------------------------
template <int NT>
__global__ void __launch_bounds__(256)
gemm_nt_kernel(const float* A, int lda, const float* Bt, int ldb,
               float* C, int ldc, const float* bias,
               int M, int N, int K, int mode, int lowerOnly) {
  const int wavesPerBlock = blockDim.x >> 5;
  int wave = blockIdx.x * wavesPerBlock + (threadIdx.x >> 5);
  int lane = threadIdx.x & 31;
  int nStrips = N / (NT * 16);
  int mStrips = M >> 4;
  if (wave >= mStrips * nStrips) return;            // wave-uniform
  int sm = wave / nStrips;
  int sn = wave - sm * nStrips;
  int r0 = sm << 4, c0 = sn * (NT * 16);
  if (lowerOnly && c0 > r0 + 15) return;            // wave-uniform

  int l16 = lane & 15, lhi = lane >> 4;
  const float* aR = A  + (r0 + l16) * lda + (lhi << 1);
  const float* bR = Bt + (c0 + l16) * ldb + (lhi << 1);
  const int ldb16 = ldb << 4;

  const v8f vz = {0.f, 0.f, 0.f, 0.f, 0.f, 0.f, 0.f, 0.f};
  v8f acc[NT];
#pragma unroll
  for (int n = 0; n < NT; ++n) acc[n] = vz;

  // software pipeline: prefetch next k-fragments before consuming current
  v2f a = *(const v2f*)(aR);
  v2f b[NT];
#pragma unroll
  for (int n = 0; n < NT; ++n) b[n] = *(const v2f*)(bR + n * ldb16);

  for (int k = 4; k < K; k += 4) {
    v2f a2 = *(const v2f*)(aR + k);
    v2f b2[NT];
#pragma unroll
    for (int n = 0; n < NT; ++n) b2[n] = *(const v2f*)(bR + n * ldb16 + k);
#pragma unroll
    for (int n = 0; n < NT; ++n) acc[n] = wmma4(a, b[n], acc[n]);
    a = a2;
#pragma unroll
    for (int n = 0; n < NT; ++n) b[n] = b2[n];
  }
#pragma unroll
  for (int n = 0; n < NT; ++n) acc[n] = wmma4(a, b[n], acc[n]);

  float bv[NT];
  if (mode == 3) {
#pragma unroll
    for (int n = 0; n < NT; ++n) bv[n] = bias[c0 + n * 16 + l16];
  }
  int rowBase = r0 + (lhi << 3);
#pragma unroll
  for (int r = 0; r < 8; ++r) {
    float* cp = C + (rowBase + r) * ldc + c0 + l16;
#pragma unroll
    for (int n = 0; n < NT; ++n) {
      float v = acc[n][r];
      if (mode == 0)      cp[n * 16] = v;
      else if (mode == 1) cp[n * 16] -= v;
      else if (mode == 2) cp[n * 16] += v;
      else                cp[n * 16] = v + bv[n];
    }
  }
}

// ---------------------------------------------------------------------------
// LDS-staged SYRK: trailing(RxR) -= panel(Rx128) @ panel^T, lower tiles only.
// Block = 256 threads computes one 128x128 C tile. Panel tiles are staged
// global->LDS with CDNA5 async-to-LDS copies (ASYNCcnt), then 8 waves each
// compute a 16x128 strip from LDS (conflict-free ds_load_b64 fragments).
// ---------------------------------------------------------------------------
__global__ void __launch_bounds__(256)
syrk_lds_kernel(const float* panel, float* Ct, int R) {
  int tc = blockIdx.x, tr = blockIdx.y;
  if (tc > tr) return;                              // block-uniform, pre-barrier
  __shared__ float As[NB * LP];
  __shared__ float Bs[NB * LP];
  int t = threadIdx.x;

  const float* Ag = panel + tr * NB * NY;
  const float* Bg = panel + tc * NB * NY;
#pragma unroll
  for (int i = 0; i < 16; ++i) {
    int idx = t + i * 256;              // 0..4095 -> 128 rows x 32 b128-chunks
    int row = idx >> 5;
    int cq  = (idx & 31) << 2;          // float offset, 16B aligned
    const float* ga = Ag + row * NY + cq;
    const float* gb = Bg + row * NY + cq;
    unsigned la = (unsigned)(uintptr_t)&As[row * LP + cq];  // LDS byte offset
    unsigned lb = (unsigned)(uintptr_t)&Bs[row * LP + cq];
    asm volatile("global_load_async_to_lds_b128 %0, %1, off"
                 :: "v"(la), "v"(ga) : "memory");
    asm volatile("global_load_async_to_lds_b128 %0, %1, off"
                 :: "v"(lb), "v"(gb) : "memory");
  }
  asm volatile("s_wait_asynccnt 0x0" ::: "memory");
  __syncthreads();

  int wv = t >> 5, lane = t & 31;
  int l16 = lane & 15, lhi = lane >> 4;
  const float* aL = &As[(wv * 16 + l16) * LP + (lhi << 1)];
  const float* bL = &Bs[l16 * LP + (lhi << 1)];

  const v8f vz = {0.f, 0.f, 0.f, 0.f, 0.f, 0.f, 0.f, 0.f};
  v8f acc[8];
#pragma unroll
  for (int n = 0; n < 8; ++n) acc[n] = vz;

#pragma unroll 4
  for (int k = 0; k < NB; k += 4) {
    v2f a = *(const v2f*)(aL + k);
#pragma unroll
    for (int n = 0; n < 8; ++n) {
      v2f b = *(const v2f*)(bL + n * 16 * LP + k);
      acc[n] = wmma4(a, b, acc[n]);
    }
  }

  int rowBase = tr * NB + wv * 16 + (lhi << 3);
  int colBase = tc * NB + l16;
#pragma unroll
  for (int r = 0; r < 8; ++r) {
    float* cp = Ct + (rowBase + r) * NY + colBase;
#pragma unroll
    for (int n = 0; n < 8; ++n) cp[n * 16] -= acc[n][r];
  }
}

// ---------------------------------------------------------------------------
// prep: Az = tril(Lz) tril(Lz)^T + jI ; Cz = chol(Az);
//       Y = Cz^{-1} tril(Lz) ; M = I - Y^T Y.   One block of 64 threads.
// ---------------------------------------------------------------------------
__global__ void prep_kernel(const float* Lz, float* Cz, float* Yo, float* Mo) {
  __shared__ float A[64][65];
  __shared__ float Ys[64][65];
  int i = threadIdx.x;

  for (int j = 0; j <= i; ++j) {
    float s = (i == j) ? JITTERF : 0.f;
    for (int k = 0; k <= j; ++k) s += Lz[i * 64 + k] * Lz[j * 64 + k];
    A[i][j] = s;
  }
  __syncthreads();
  for (int k = 0; k < 64; ++k) {
    if (i == k) A[k][k] = sqrtf(A[k][k]);
    __syncthreads();
    if (i > k) A[i][k] /= A[k][k];
    __syncthreads();
    if (i > k) {
      float aik = A[i][k];
      for (int j = k + 1; j <= i; ++j) A[i][j] -= aik * A[j][k];
    }
    __syncthreads();
  }
  for (int j = 0; j < 64; ++j) Cz[i * 64 + j] = (j <= i) ? A[i][j] : 0.f;

  for (int ii = 0; ii < 64; ++ii) {       // column i of Y: Cz y = tril(Lz)[:,i]
    float s = (ii >= i) ? Lz[ii * 64 + i] : 0.f;
    for (int p = 0; p < ii; ++p) s -= A[ii][p] * Ys[p][i];
    Ys[ii][i] = s / A[ii][ii];
  }
  __syncthreads();
  for (int j = 0; j < 64; ++j) Yo[i * 64 + j] = Ys[i][j];
  for (int b = 0; b < 64; ++b) {
    float s = (i == b) ? 1.f : 0.f;
    for (int ii = 0; ii < 64; ++ii) s -= Ys[ii][i] * Ys[ii][b];
    Mo[i * 64 + b] = s;
  }
}

// ---------------------------------------------------------------------------
// potf2 (NB=128): in-place Cholesky of a 128x128 diagonal block (lower) +
// its triangular inverse into `inv` (row-major, upper zeroed). 128 threads.
// ---------------------------------------------------------------------------
__global__ void potf2_kernel(float* D, int ld, float* inv) {
  __shared__ float A[NB][NB + 1];
  __shared__ float Xi[NB][NB + 1];
  int i = threadIdx.x;

  for (int j = 0; j <= i; ++j) A[i][j] = D[i * ld + j];
  __syncthreads();
  for (int k = 0; k < NB; ++k) {
    if (i == k) A[k][k] = sqrtf(A[k][k]);
    __syncthreads();
    if (i > k) A[i][k] /= A[k][k];
    __syncthreads();
    if (i > k) {
      float aik = A[i][k];
      for (int j = k + 1; j <= i; ++j) A[i][j] -= aik * A[j][k];
    }
    __syncthreads();
  }
  for (int j = 0; j <= i; ++j) D[i * ld + j] = A[i][j];

  for (int ii = 0; ii < NB; ++ii) Xi[ii][i] = 0.f;   // inv column i: L x = e_i
  for (int ii = i; ii < NB; ++ii) {
    float s = (ii == i) ? 1.f : 0.f;
    for (int p = i; p < ii; ++p) s -= A[ii][p] * Xi[p][i];
    Xi[ii][i] = s / A[ii][ii];
  }
  __syncthreads();
  for (int j = 0; j < NB; ++j) inv[i * NB + j] = Xi[i][j];
}

// S diag 16x16 blocks += tril(Ly_n) tril(Ly_n)^T + jI
__global__ void diag_fix_kernel(const float* Ly, float* S) {
  int n = blockIdx.x, t = threadIdx.x;
  int i = t >> 4, j = t & 15;
  const float* Lyn = Ly + n * 256;
  float s = (i == j) ? JITTERF : 0.f;
  int km = (i < j) ? i : j;
  for (int k = 0; k <= km; ++k) s += Lyn[i * 16 + k] * Lyn[j * 16 + k];
  S[(n * 16 + i) * NY + (n * 16 + j)] += s;
}

__global__ void zero_upper_kernel(float* S) {
  int col = blockIdx.x * blockDim.x + threadIdx.x;
  int row = blockIdx.y;
  if (col > row) S[row * NY + col] = 0.f;
}

// ---------------------------------------------------------------------------
template <int NT>
static inline void launch_gemm(hipStream_t s, int mode, int lowerOnly,
                               const float* A, int lda, const float* Bt, int ldb,
                               float* C, int ldc, const float* bias,
                               int M, int N, int K) {
  int strips = (M >> 4) * (N / (NT * 16));
  int blocks = (strips + 7) >> 3;   // 8 waves per 256-thread block
  gemm_nt_kernel<NT><<<blocks, 256, 0, s>>>(A, lda, Bt, ldb, C, ldc, bias,
                                            M, N, K, mode, lowerOnly);
}

extern "C" void kernel_launch(void* const* d_in, const int* in_sizes, int n_in,
                              void* d_out, int out_size, void* d_ws, size_t ws_size,
                              hipStream_t stream) {
  (void)in_sizes; (void)n_in; (void)out_size; (void)ws_size;
  const float* m   = (const float*)d_in[0];   // 8256
  const float* Lz  = (const float*)d_in[1];   // 64x64
  const float* Ly  = (const float*)d_in[2];   // 512x16x16
  const float* Lyz = (const float*)d_in[3];   // 8192x64
  const float* eps = (const float*)d_in[4];   // 128x8256
  float* out = (float*)d_out;                 // 128x8256

  char* ws = (char*)d_ws;
  size_t off = 0;
  float* S   = (float*)(ws + off); off += (size_t)NY * NY * 4;   // 256 MB
  float* Cyz = (float*)(ws + off); off += (size_t)NY * 64 * 4;
  float* Rm  = (float*)(ws + off); off += (size_t)NY * 64 * 4;
  float* Cz  = (float*)(ws + off); off += 64 * 64 * 4;
  float* Yb  = (float*)(ws + off); off += 64 * 64 * 4;
  float* Mb  = (float*)(ws + off); off += 64 * 64 * 4;
  float* inv = (float*)(ws + off); off += NB * NB * 4;

  // Phase 0: small factors (Cz, Y, M)
  prep_kernel<<<1, 64, 0, stream>>>(Lz, Cz, Yb, Mb);

  // Phase 1: Cyz = Lyz @ Y^T
  launch_gemm<4>(stream, 0, 0, Lyz, 64, Yb, 64, Cyz, 64, nullptr, NY, 64, 64);

  // Phase 2: S = (Lyz @ M) @ Lyz^T + blockdiag(tril(Ly)tril(Ly)^T) + jI
  launch_gemm<4>(stream, 0, 0, Lyz, 64, Mb, 64, Rm, 64, nullptr, NY, 64, 64);
  launch_gemm<4>(stream, 0, 0, Rm, 64, Lyz, 64, S, NY, nullptr, NY, NY, 64);
  diag_fix_kernel<<<NBLK, 256, 0, stream>>>(Ly, S);

  // Phase 3: blocked in-place Cholesky of S, NB = 128
  for (int j = 0; j < NY / NB; ++j) {
    int o = NB * j;
    int R = NY - o - NB;
    potf2_kernel<<<1, NB, 0, stream>>>(S + o * NY + o, NY, inv);
    if (R > 0) {
      float* panel = S + (o + NB) * NY + o;
      // TRSM: panel = panel @ inv(L11)^T, in place (wave owns 16x128 strip)
      launch_gemm<8>(stream, 0, 0, panel, NY, inv, NB, panel, NY, nullptr,
                     R, NB, NB);
      // SYRK: trailing -= panel @ panel^T (lower 128x128 tiles, LDS-staged)
      syrk_lds_kernel<<<dim3(R / NB, R / NB), 256, 0, stream>>>(
          panel, S + (o + NB) * NY + (o + NB), R);
    }
  }
  zero_upper_kernel<<<dim3(NY / 256, NY), 256, 0, stream>>>(S);

  // Phase 4: out = m + eps @ chol^T
  launch_gemm<4>(stream, 3, 0, eps, DTOT, Cz, 64, out, DTOT, m,
                 NSAMPLE, 64, 64);
  launch_gemm<4>(stream, 3, 0, eps + 64, DTOT, S, NY, out + 64, DTOT, m + 64,
                 NSAMPLE, NY, NY);
  launch_gemm<4>(stream, 2, 0, eps, DTOT, Cyz, 64, out + 64, DTOT, nullptr,
                 NSAMPLE, NY, 64);
}